// Scene_73727408603813
// MI455X (gfx1250) — compile-verified
//
#include <hip/hip_runtime.h>
#include <stdint.h>

// Problem constants (must match reference).
constexpr int kNSrc    = 128;
constexpr int kC       = 5;
constexpr int kH       = 4096;
constexpr int kW       = 4096;
constexpr int kPH      = 256;
constexpr int kPW      = 256;
constexpr int kTile    = 128;   // canvas tile edge; kH,kW divisible by kTile
constexpr int kThreads = 256;   // 8 waves (wave32)

__device__ __forceinline__ int imin(int a, int b) { return a < b ? a : b; }
__device__ __forceinline__ int imax(int a, int b) { return a > b ? a : b; }

// CDNA5 async LDS->global store (ASYNCcnt-tracked). Each active lane copies
// 16 bytes from its LDS address to its global address. The low 32 bits of a
// generic pointer to __shared__ memory are the LDS byte offset, so we can
// feed the instruction with plain integer VGPR operands. TH_STORE_NT: the
// canvas writeback is write-once/never-re-read and larger than L2, so keep
// it out of the cache hierarchy and leave L2 for the patch reads.
__device__ __forceinline__ void async_store_b128_from_lds_nt(void* gaddr, const void* lds_ptr) {
#if defined(__AMDGCN__)
  unsigned long long g = (unsigned long long)(uintptr_t)gaddr;
  unsigned           l = (unsigned)(uintptr_t)lds_ptr;  // LDS byte offset
  asm volatile("global_store_async_from_lds_b128 %0, %1, off th:TH_STORE_NT"
               :
               : "v"(g), "v"(l)
               : "memory");
#else
  (void)gaddr; (void)lds_ptr;
#endif
}

__device__ __forceinline__ void wait_asynccnt0() {
#if defined(__AMDGCN__)
#if __has_builtin(__builtin_amdgcn_s_wait_asynccnt)
  __builtin_amdgcn_s_wait_asynccnt(0);
#else
  asm volatile("s_wait_asynccnt 0" ::: "memory");
#endif
#endif
}

// One block owns a 128x128 tile of one band. Tiles partition the canvas, so
// every output pixel is written exactly once (zeros where nothing overlaps):
// no global atomics, no separate zeroing pass, single trip over the data.
__global__ __launch_bounds__(kThreads) void scene_accum_kernel(
    const float* __restrict__ patches,  // [kNSrc][kC][kPH][kPW]
    const int*  __restrict__ starts,    // [kNSrc][2] (y, x)
    float* __restrict__ out) {          // [kC][kH][kW]
  __shared__ float tile[kTile * kTile];  // 64 KB LDS accumulator

  const int tx0 = blockIdx.x * kTile;
  const int ty0 = blockIdx.y * kTile;
  const int c   = blockIdx.z;
  const int tid = threadIdx.x;

  // Zero the LDS accumulator tile with b128 stores (16 per thread).
  {
    float4* t4 = (float4*)tile;
    const float4 z = make_float4(0.f, 0.f, 0.f, 0.f);
#pragma unroll
    for (int k = 0; k < (kTile * kTile / 4) / kThreads; ++k)
      t4[tid + k * kThreads] = z;
  }
  __syncthreads();

  const int col = tid & (kTile - 1);  // 0..127: column within intersection
  const int r0  = tid >> 7;           // 0..1  : row phase (256 thr / 128 cols)

  // Accumulate every patch intersecting this tile.
  for (int n = 0; n < kNSrc; ++n) {
    const int sy = starts[2 * n + 0];
    const int sx = starts[2 * n + 1];
    const int iy0 = imax(ty0, sy);
    const int iy1 = imin(ty0 + kTile, sy + kPH);
    const int ix0 = imax(tx0, sx);
    const int ix1 = imin(tx0 + kTile, sx + kPW);
    if (iy0 >= iy1 || ix0 >= ix1) continue;     // uniform skip: no overlap
    if (col >= ix1 - ix0) continue;             // lane outside intersection width

    const float* __restrict__ p =
        patches + (((size_t)n * kC + c) * kPH + (iy0 - sy)) * (size_t)kPW + (ix0 - sx);
    float* trow = &tile[(iy0 - ty0) * kTile + (ix0 - tx0)];
    const int hgt = iy1 - iy0;

    for (int r = r0; r < hgt; r += kThreads / kTile) {
      // Coalesced patch-row read; ds_add_f32 into the LDS tile (no races,
      // no per-patch barriers needed).
      atomicAdd(&trow[r * kTile + col], p[(size_t)r * kPW + col]);
    }
  }
  __syncthreads();

  // Bulk writeback: 64 KB LDS tile -> global via async b128 stores.
  // 4096 float4s / 256 threads = 16 async stores per thread; each 32-lane
  // wave iteration covers one contiguous 512 B canvas-row segment.
  float* obase = out + ((size_t)c * kH + ty0) * (size_t)kW + tx0;
#pragma unroll
  for (int k = 0; k < (kTile * kTile / 4) / kThreads; ++k) {
    const int f   = tid + k * kThreads;  // float4 index within tile
    const int row = f >> 5;              // 32 float4 per 128-float row
    const int q   = (f & 31) * 4;        // float offset within row
    async_store_b128_from_lds_nt(obase + (size_t)row * kW + q, &tile[row * kTile + q]);
  }
  wait_asynccnt0();
}

extern "C" void kernel_launch(void* const* d_in, const int* in_sizes, int n_in,
                              void* d_out, int out_size, void* d_ws, size_t ws_size,
                              hipStream_t stream) {
  (void)in_sizes; (void)n_in; (void)out_size; (void)d_ws; (void)ws_size;
  const float* patches = (const float*)d_in[0];
  const int*   starts  = (const int*)d_in[1];
  float*       out     = (float*)d_out;

  dim3 grid(kW / kTile, kH / kTile, kC);  // 32 x 32 x 5 = 5120 blocks
  scene_accum_kernel<<<grid, kThreads, 0, stream>>>(patches, starts, out);
}